// Net1_71038759076121
// MI455X (gfx1250) — compile-verified
//
#include <hip/hip_runtime.h>

typedef __attribute__((ext_vector_type(2))) float v2f;
typedef __attribute__((ext_vector_type(4))) float v4f;
typedef __attribute__((ext_vector_type(8))) float v8f;

#define D 16

// ---------------- kernel 0: zero the workspace (agg + cnt) ----------------
__global__ void zero_f32(float* __restrict__ p, long n) {
    long i      = (long)blockIdx.x * blockDim.x + threadIdx.x;
    long stride = (long)gridDim.x * blockDim.x;
    long n4     = n >> 2;
    v4f z = {0.f, 0.f, 0.f, 0.f};
    for (long j = i; j < n4; j += stride)
        ((v4f*)p)[j] = z;
    long tail = n & 3;
    if (i < tail) p[(n4 << 2) + i] = 0.f;
}

// ---------------- kernel 1: edge scatter (mean-agg numerator + counts) ----
// 16 lanes cooperate on one edge: lanes read x[src][0..15] (one coalesced
// 64B line) and issue 16 contiguous global_atomic_add_f32 into agg[dst].
// Lane 0 bumps the degree counter. Atomics are non-returning (STOREcnt only).
__global__ void sage_scatter(const float* __restrict__ x,
                             const int*   __restrict__ ei,
                             float* __restrict__ agg,
                             float* __restrict__ cnt,
                             int nEdges) {
    long gid  = (long)blockIdx.x * blockDim.x + threadIdx.x;
    long e    = gid >> 4;
    int  lane = (int)(gid & 15);
    if (e >= nEdges) return;
    int src = ei[e];                   // edge_index[0][e]
    int dst = ei[(long)nEdges + e];    // edge_index[1][e]
    float v = x[(long)src * D + lane];
    atomicAdd(&agg[(long)dst * D + lane], v);
    if (lane == 0) atomicAdd(&cnt[dst], 1.0f);
}

// ---------------- kernel 2: WMMA transform -------------------------------
// One wave32 per 16-node tile. OUT[16x16] = MEAN[16x16]@W_l^T + X[16x16]@W_r^T + b_l
// computed as 4 K-chunks of V_WMMA_F32_16X16X4_F32 per input matrix (8 WMMAs),
// all chained through one f32 accumulator seeded with the bias.
//
// A layout (16x4 f32, ISA 7.12.2): lane L holds row m=L%16; VGPR0 = K={0|2},
// VGPR1 = K={1|3} selected by L/16  ->  a = float2 at A[m][4c + 2*(L>>4)].
// B layout (4x16): lane L holds col n=L%16 with the same K striping ->
// b = float2 at W[n][4c + 2*(L>>4)]  (W stored [out][in] row-major).
// C/D layout: VGPR r, lane L -> (row r + 8*(L>>4), col L%16).
__global__ void sage_transform(const float* __restrict__ x,
                               const float* __restrict__ agg,
                               const float* __restrict__ cnt,
                               const float* __restrict__ Wl,
                               const float* __restrict__ bl,
                               const float* __restrict__ Wr,
                               float* __restrict__ out,
                               int nTiles) {
    int wave = threadIdx.x >> 5;
    int lane = threadIdx.x & 31;
    int tile = blockIdx.x * (blockDim.x >> 5) + wave;
    if (tile >= nTiles) return;        // wave-uniform: EXEC stays all-1s

    int  m16  = lane & 15;             // A row (node) == B col (out feature)
    int  hi   = lane >> 4;             // selects K+2 half
    long node = (long)tile * 16 + m16;

    float inv  = 1.0f / fmaxf(cnt[node], 1.0f);
    float bias = bl[m16];

    v8f c;
#pragma unroll
    for (int r = 0; r < 8; ++r) c[r] = bias;

#pragma unroll
    for (int k = 0; k < 4; ++k) {
        int k0 = 4 * k + 2 * hi;
        // mean tile chunk * W_l^T chunk
        v2f am = *(const v2f*)(agg + node * D + k0);
        am = am * inv;
        v2f bl2 = *(const v2f*)(Wl + m16 * D + k0);
        c = __builtin_amdgcn_wmma_f32_16x16x4_f32(false, am, false, bl2,
                                                  (short)0, c, false, false);
        // x tile chunk * W_r^T chunk
        v2f ax = *(const v2f*)(x + node * D + k0);
        v2f br2 = *(const v2f*)(Wr + m16 * D + k0);
        c = __builtin_amdgcn_wmma_f32_16x16x4_f32(false, ax, false, br2,
                                                  (short)0, c, false, false);
    }

#pragma unroll
    for (int r = 0; r < 8; ++r) {
        long row = (long)tile * 16 + r + 8 * hi;
        out[row * D + m16] = c[r];
    }
}

// ---------------- kernel 3: scalar tail for nNodes % 16 (unused at 500K) --
__global__ void sage_tail(const float* __restrict__ x,
                          const float* __restrict__ agg,
                          const float* __restrict__ cnt,
                          const float* __restrict__ Wl,
                          const float* __restrict__ bl,
                          const float* __restrict__ Wr,
                          float* __restrict__ out,
                          int firstNode, int nNodes) {
    int node = firstNode + blockIdx.x * blockDim.x + threadIdx.x;
    if (node >= nNodes) return;
    float inv = 1.0f / fmaxf(cnt[node], 1.0f);
    float mean[D], xv[D];
#pragma unroll
    for (int k = 0; k < D; ++k) {
        mean[k] = agg[(long)node * D + k] * inv;
        xv[k]   = x[(long)node * D + k];
    }
#pragma unroll
    for (int n = 0; n < D; ++n) {
        float acc = bl[n];
#pragma unroll
        for (int k = 0; k < D; ++k)
            acc += mean[k] * Wl[n * D + k] + xv[k] * Wr[n * D + k];
        out[(long)node * D + n] = acc;
    }
}

extern "C" void kernel_launch(void* const* d_in, const int* in_sizes, int n_in,
                              void* d_out, int out_size, void* d_ws, size_t ws_size,
                              hipStream_t stream) {
    const float* x  = (const float*)d_in[0];
    const int*   ei = (const int*)d_in[1];
    const float* Wl = (const float*)d_in[2];
    const float* bl = (const float*)d_in[3];
    const float* Wr = (const float*)d_in[4];
    float* out = (float*)d_out;

    int nNodes = in_sizes[0] / D;
    int nEdges = in_sizes[1] / 2;

    float* agg = (float*)d_ws;                    // nNodes * 16 floats
    float* cnt = agg + (size_t)nNodes * D;        // nNodes floats

    // zero agg + cnt
    long zc = (long)nNodes * (D + 1);
    zero_f32<<<2048, 256, 0, stream>>>(agg, zc);

    // scatter: 16 lanes per edge
    long threads = (long)nEdges * 16;
    int  sblocks = (int)((threads + 255) / 256);
    sage_scatter<<<sblocks, 256, 0, stream>>>(x, ei, agg, cnt, nEdges);

    // WMMA transform: 1 wave per 16-node tile, 8 waves per block
    int nTiles = nNodes / 16;
    if (nTiles > 0) {
        int tblocks = (nTiles + 7) / 8;
        sage_transform<<<tblocks, 256, 0, stream>>>(x, agg, cnt, Wl, bl, Wr,
                                                    out, nTiles);
    }
    int firstTail = nTiles * 16;
    int nTail = nNodes - firstTail;
    if (nTail > 0) {
        sage_tail<<<(nTail + 255) / 256, 256, 0, stream>>>(x, agg, cnt, Wl, bl,
                                                           Wr, out, firstTail,
                                                           nNodes);
    }
}